// KNNDist_27015344292564
// MI455X (gfx1250) — compile-verified
//
#include <hip/hip_runtime.h>
#include <math.h>

// ---------------------------------------------------------------------------
// kNN outlier loss for MI455X (gfx1250, wave32).
//
// dist(i,j) = xx_i + xx_j - 2*dot(p_i,p_j), computed as a rank-4 WMMA:
//   A row i = (-2x_i, -2y_i, -2z_i, 1)
//   B col j = ( x_j,   y_j,   z_j, xx_j)
//   => C[i][j] = -2*dot + xx_j ; then add xx_i per row, clamp at -1e-6.
// Each wave owns 32 query rows; per 16-column tile it runs two
// v_wmma_f32_16x16x4_f32 ops, spills the 32x16 tile to wave-private LDS
// (row stride padded to 20 floats so each row is 16B-aligned), reads its
// own row back with 4x ds_load_b128, and maintains a sorted top-6 smallest.
// value = mean of the 5 smallest excluding the global min (self).
// ---------------------------------------------------------------------------

typedef float v2f __attribute__((ext_vector_type(2)));
typedef float v8f __attribute__((ext_vector_type(8)));

#define ALPHA 1.05f
#define ROWS_PER_BLOCK 256   // 8 waves * 32 rows
#define CCH 1024             // column points chunked through LDS
#define TPAD 20              // tile row stride (floats): 80B, 16B-aligned

__global__ __launch_bounds__(256) void knn_value_kernel(
    const float* __restrict__ pc,   // [B, K, 3]
    float* __restrict__ value,      // [B, K]
    int K, int blocksPerBatch)
{
    const int b   = blockIdx.x / blocksPerBatch;
    const int rb0 = (blockIdx.x % blocksPerBatch) * ROWS_PER_BLOCK;
    const float* __restrict__ pcb = pc + (size_t)b * K * 3;

    __shared__ float4 srow[ROWS_PER_BLOCK];    // block's query rows (x,y,z,xx)
    __shared__ float4 scol[CCH];               // current column chunk
    __shared__ float  stile[8][32][TPAD];      // per-wave 32x16 dist tile (+pad)

    // --- load this block's 256 query points, precompute squared norms ---
    {
        const int r = rb0 + (int)threadIdx.x;
        const float x = pcb[(size_t)r * 3 + 0];
        const float y = pcb[(size_t)r * 3 + 1];
        const float z = pcb[(size_t)r * 3 + 2];
        srow[threadIdx.x] = make_float4(x, y, z, x * x + y * y + z * z);
    }
    __syncthreads();

    const int lane = threadIdx.x & 31;
    const int w    = threadIdx.x >> 5;
    const int ln   = lane & 15;       // column / M within half
    const int hi   = lane >> 4;       // 0: K=0,1 half  1: K=2,3 half
    const int rowBase = w * 32;       // local row base within block

    // --- hoist A fragments for the wave's two 16-row tiles ---
    v2f a0, a1;
    {
        const float4 p0 = srow[rowBase + ln];
        const float4 p1 = srow[rowBase + 16 + ln];
        if (hi == 0) {
            a0.x = -2.0f * p0.x; a0.y = -2.0f * p0.y;   // K=0,1
            a1.x = -2.0f * p1.x; a1.y = -2.0f * p1.y;
        } else {
            a0.x = -2.0f * p0.z; a0.y = 1.0f;           // K=2,3
            a1.x = -2.0f * p1.z; a1.y = 1.0f;
        }
    }
    // row norms matching the C/D accumulator layout (VGPR i -> row i + 8*hi)
    float xx0[8], xx1[8];
#pragma unroll
    for (int i = 0; i < 8; ++i) {
        xx0[i] = srow[rowBase + i + 8 * hi].w;
        xx1[i] = srow[rowBase + 16 + i + 8 * hi].w;
    }

    // sorted ascending top-6 smallest distances for this lane's row
    float t0 = 3.0e38f, t1 = 3.0e38f, t2 = 3.0e38f,
          t3 = 3.0e38f, t4 = 3.0e38f, t5 = 3.0e38f;

    // this lane's read-back row (16B-aligned: stride 80B)
    const float4* __restrict__ myrow = (const float4*)&stile[w][lane][0];

#define KNN_INSERT(VV)                                              \
    {                                                               \
        const float xv = (VV);                                      \
        if (xv < t5) {                                              \
            t5 = xv;                                                \
            if (t5 < t4) { float s = t4; t4 = t5; t5 = s; }         \
            if (t4 < t3) { float s = t3; t3 = t4; t4 = s; }         \
            if (t3 < t2) { float s = t2; t2 = t3; t3 = s; }         \
            if (t2 < t1) { float s = t1; t1 = t2; t2 = s; }         \
            if (t1 < t0) { float s = t0; t0 = t1; t1 = s; }         \
        }                                                           \
    }

    for (int cb = 0; cb < K; cb += CCH) {
        __syncthreads();   // previous chunk fully consumed before overwrite
        for (int idx = threadIdx.x; idx < CCH; idx += blockDim.x) {
            const int cg = cb + idx;
            const float x = pcb[(size_t)cg * 3 + 0];
            const float y = pcb[(size_t)cg * 3 + 1];
            const float z = pcb[(size_t)cg * 3 + 2];
            scol[idx] = make_float4(x, y, z, x * x + y * y + z * z);
        }
        __syncthreads();

        for (int ct = 0; ct < CCH; ct += 16) {
            // B fragment: lanes 0-15 carry (x,y) = K rows 0,1; 16-31 carry (z,xx)
            const float4 p = scol[ct + ln];
            v2f bf;
            bf.x = hi ? p.z : p.x;
            bf.y = hi ? p.w : p.y;

            v8f c0 = {};
            v8f c1 = {};
            c0 = __builtin_amdgcn_wmma_f32_16x16x4_f32(
                false, a0, false, bf, (short)0, c0, false, false);
            c1 = __builtin_amdgcn_wmma_f32_16x16x4_f32(
                false, a1, false, bf, (short)0, c1, false, false);

            // spill 32x16 tile: add per-row xx, clamp like jnp.clip(a_min=-1e-6)
#pragma unroll
            for (int i = 0; i < 8; ++i) {
                stile[w][i + 8 * hi][ln]      = fmaxf(c0[i] + xx0[i], -1e-6f);
                stile[w][16 + i + 8 * hi][ln] = fmaxf(c1[i] + xx1[i], -1e-6f);
            }

            // lane <lane> owns local row <lane>: fetch its 16 columns with
            // 4x b128 LDS loads (one wait), then insert from registers.
            const float4 q0 = myrow[0];
            const float4 q1 = myrow[1];
            const float4 q2 = myrow[2];
            const float4 q3 = myrow[3];
            KNN_INSERT(q0.x); KNN_INSERT(q0.y); KNN_INSERT(q0.z); KNN_INSERT(q0.w);
            KNN_INSERT(q1.x); KNN_INSERT(q1.y); KNN_INSERT(q1.z); KNN_INSERT(q1.w);
            KNN_INSERT(q2.x); KNN_INSERT(q2.y); KNN_INSERT(q2.z); KNN_INSERT(q2.w);
            KNN_INSERT(q3.x); KNN_INSERT(q3.y); KNN_INSERT(q3.z); KNN_INSERT(q3.w);
        }
    }
#undef KNN_INSERT

    // drop the smallest (self ~0), mean of remaining 5 kNN distances
    const float val = (t1 + t2 + t3 + t4 + t5) * 0.2f;
    value[(size_t)b * K + rb0 + rowBase + lane] = val;
}

__global__ __launch_bounds__(256) void knn_loss_kernel(
    const float* __restrict__ value,    // [B, K]
    const float* __restrict__ weights,  // [B]
    float* __restrict__ loss,           // [B]
    int K)
{
    const int b = blockIdx.x;
    const float* __restrict__ v = value + (size_t)b * K;

    __shared__ float ssum[256];
    __shared__ float ssq[256];
    __shared__ float thr_s;

    float s = 0.0f, sq = 0.0f;
    for (int i = threadIdx.x; i < K; i += blockDim.x) {
        const float x = v[i];
        s += x; sq += x * x;
    }
    ssum[threadIdx.x] = s;
    ssq[threadIdx.x]  = sq;
    for (int off = 128; off > 0; off >>= 1) {
        __syncthreads();
        if ((int)threadIdx.x < off) {
            ssum[threadIdx.x] += ssum[threadIdx.x + off];
            ssq[threadIdx.x]  += ssq[threadIdx.x + off];
        }
    }
    __syncthreads();
    if (threadIdx.x == 0) {
        const float mean = ssum[0] / (float)K;
        const float var  = (ssq[0] - (float)K * mean * mean) / (float)(K - 1);
        thr_s = mean + ALPHA * sqrtf(fmaxf(var, 0.0f));   // unbiased std (ddof=1)
    }
    __syncthreads();
    const float thr = thr_s;

    float m = 0.0f;
    for (int i = threadIdx.x; i < K; i += blockDim.x) {
        const float x = v[i];
        if (x > thr) m += x;
    }
    ssum[threadIdx.x] = m;
    for (int off = 128; off > 0; off >>= 1) {
        __syncthreads();
        if ((int)threadIdx.x < off) ssum[threadIdx.x] += ssum[threadIdx.x + off];
    }
    __syncthreads();
    if (threadIdx.x == 0)
        loss[b] = (ssum[0] / (float)K) * weights[b];
}

__global__ void knn_final_kernel(const float* __restrict__ loss,
                                 float* __restrict__ out, int B)
{
    if (threadIdx.x == 0) {
        float s = 0.0f;
        for (int b = 0; b < B; ++b) s += loss[b];
        out[0] = s / (float)B;
    }
}

extern "C" void kernel_launch(void* const* d_in, const int* in_sizes, int n_in,
                              void* d_out, int out_size, void* d_ws, size_t ws_size,
                              hipStream_t stream)
{
    const float* pc      = (const float*)d_in[0];   // [B,K,3] fp32
    const float* weights = (const float*)d_in[1];   // [B] fp32
    float* out = (float*)d_out;                     // scalar fp32

    const int B = in_sizes[1];                      // 16
    const int K = in_sizes[0] / (3 * B);            // 4096

    float* ws_value = (float*)d_ws;                 // B*K floats
    float* ws_loss  = ws_value + (size_t)B * K;     // B floats

    const int blocksPerBatch = K / ROWS_PER_BLOCK;  // 16
    knn_value_kernel<<<dim3(B * blocksPerBatch), dim3(256), 0, stream>>>(
        pc, ws_value, K, blocksPerBatch);
    knn_loss_kernel<<<dim3(B), dim3(256), 0, stream>>>(ws_value, weights, ws_loss, K);
    knn_final_kernel<<<dim3(1), dim3(32), 0, stream>>>(ws_loss, out, B);
}